// point_OHEM_loss_23536420782207
// MI455X (gfx1250) — compile-verified
//
#include <hip/hip_runtime.h>
#include <hip/hip_bf16.h>
#include <stdint.h>

#define NPIX   (512 * 512)   // pixels per sample
#define BATCH  8
#define EPS_F  1e-6f
#define EPSSQ  1e-12f
#define RADIO  0.7f

#define K1_T   256
#define SEL_T  1024
#define CHUNK  4096          // floats per TDM chunk = 16 KB

typedef unsigned int u32x4 __attribute__((ext_vector_type(4)));
typedef int          i32x8 __attribute__((ext_vector_type(8)));
typedef int          i32x4 __attribute__((ext_vector_type(4)));

// ---------------------------------------------------------------------------
// TDM helpers (gfx1250 Tensor Data Mover). Descriptor per CDNA5 ISA §8:
//   group0: count=1 | lds_addr | global_addr[56:0] | type=2
//   group1: data_size=4B, tensor_dim0 = tile_dim0 = elems, tensor_dim1 = tile_dim1 = 1
// One wave issues the DMA; completion tracked with TENSORcnt.
// This toolchain (clang-23 / therock-10.0) uses the 6-arg builtin:
//   (uint32x4 g0, int32x8 g1, int32x4, int32x4, int32x8, i32 cpol)
// ---------------------------------------------------------------------------
__device__ __forceinline__ unsigned lds_byte_offset(const void* p) {
    // generic pointers into LDS carry the wave-relative LDS byte offset in the low 32 bits
    return (unsigned)(uintptr_t)p;
}

__device__ __forceinline__ void tdm_load_f32_chunk(const float* gsrc, unsigned lds_off, int elems) {
#if defined(__gfx1250__)
    unsigned long long ga = (unsigned long long)(uintptr_t)gsrc;
    u32x4 g0;
    g0.x = 1u;                                                   // count=1 (valid descriptor)
    g0.y = lds_off;                                              // lds_addr (bytes)
    g0.z = (unsigned)(ga & 0xffffffffull);                       // global_addr[31:0]
    g0.w = (unsigned)((ga >> 32) & 0x01ffffffull) | (2u << 30);  // global_addr[56:32], type=2
    i32x8 g1;
    g1[0] = (int)(2u << 16);                                     // data_size=2 -> 4 bytes
    g1[1] = (int)(((unsigned)elems & 0xffffu) << 16);            // tensor_dim0[15:0]
    g1[2] = (int)((((unsigned)elems >> 16) & 0xffffu) | (1u << 16)); // tensor_dim0[31:16], tensor_dim1=1
    g1[3] = (int)(((unsigned)elems & 0xffffu) << 16);            // tile_dim0 = elems
    g1[4] = 1;                                                   // tile_dim1 = 1 (tile_dim2 unused)
    g1[5] = elems;                                               // tensor_dim0_stride (low 32)
    g1[6] = 0;
    g1[7] = 0;
    i32x4 gz4 = {0, 0, 0, 0};
    i32x8 gz8 = {0, 0, 0, 0, 0, 0, 0, 0};
    __builtin_amdgcn_tensor_load_to_lds(g0, g1, gz4, gz4, gz8, 0);
#else
    (void)gsrc; (void)lds_off; (void)elems;
#endif
}

__device__ __forceinline__ void tdm_wait_all() {
#if defined(__gfx1250__)
    __builtin_amdgcn_s_wait_tensorcnt(0);
#endif
}

// ---------------------------------------------------------------------------
// Kernel 0: zero the per-sample unknown-pixel counters (ws is not re-poisoned
// between replays, so we must reinitialize every call).
// ---------------------------------------------------------------------------
__global__ void ohem_init_kernel(int* __restrict__ counts) {
    if (threadIdx.x < BATCH) counts[threadIdx.x] = 0;
}

// ---------------------------------------------------------------------------
// Kernel 1: fused elementwise pass, 4 pixels/thread with B128 loads/stores.
//   dA[b,n] = sqrt(((alpha/255 - pred) * unknown)^2 + eps^2)
//   dC[b,n] = sum_c sqrt(((image_c - (fg_c*p + (1-p)*bg_c)) * unknown)^2 + eps^2)
// Per-sample unknown count: per-thread count -> wave32 shuffle reduce ->
// one integer atomic per wave (deterministic).
// Bandwidth bound: ~101 MB in / 16.8 MB out -> ~5 us at 23.3 TB/s.
// ---------------------------------------------------------------------------
__global__ __launch_bounds__(K1_T) void ohem_diff_kernel(
    const float4* __restrict__ image, const float4* __restrict__ alpha,
    const float4* __restrict__ pred,  const int4*   __restrict__ trimap,
    const float4* __restrict__ fg,    const float4* __restrict__ bg,
    float4* __restrict__ dA, float4* __restrict__ dC, int* __restrict__ counts)
{
    const size_t g  = (size_t)blockIdx.x * K1_T + threadIdx.x;   // vec4 index < BATCH*NPIX/4
    const int    b  = (int)(g >> 16);                            // NPIX/4 == 2^16
    const size_t n4 = g & (size_t)(NPIX / 4 - 1);                // vec4 index within sample

    const int4   tri = trimap[g];
    const float4 al  = alpha[g];
    const float4 pr  = pred[g];

    float u[4], p[4];
    u[0] = (tri.x == 128) ? 1.0f : 0.0f;  u[1] = (tri.y == 128) ? 1.0f : 0.0f;
    u[2] = (tri.z == 128) ? 1.0f : 0.0f;  u[3] = (tri.w == 128) ? 1.0f : 0.0f;
    p[0] = pr.x; p[1] = pr.y; p[2] = pr.z; p[3] = pr.w;

    float da[4] = {
        (al.x * (1.0f / 255.0f) - p[0]) * u[0],
        (al.y * (1.0f / 255.0f) - p[1]) * u[1],
        (al.z * (1.0f / 255.0f) - p[2]) * u[2],
        (al.w * (1.0f / 255.0f) - p[3]) * u[3]};
#pragma unroll
    for (int j = 0; j < 4; ++j) da[j] = sqrtf(da[j] * da[j] + EPSSQ);

    const size_t cbase = (size_t)b * 3 * (NPIX / 4) + n4;
    float s[4] = {0.0f, 0.0f, 0.0f, 0.0f};
#pragma unroll
    for (int c = 0; c < 3; ++c) {
        const size_t o  = cbase + (size_t)c * (NPIX / 4);
        const float4 im = image[o], f = fg[o], gb = bg[o];
        const float imv[4] = {im.x, im.y, im.z, im.w};
        const float fv[4]  = {f.x,  f.y,  f.z,  f.w};
        const float gv[4]  = {gb.x, gb.y, gb.z, gb.w};
#pragma unroll
        for (int j = 0; j < 4; ++j) {
            float e = (imv[j] - (fv[j] * p[j] + (1.0f - p[j]) * gv[j])) * u[j];
            s[j] += sqrtf(e * e + EPSSQ);
        }
    }
    dA[g] = make_float4(da[0], da[1], da[2], da[3]);
    dC[g] = make_float4(s[0],  s[1],  s[2],  s[3]);

    // per-wave unknown count (waves never straddle a sample: NPIX/4 % 32 == 0)
    int cnt = (int)(u[0] + u[1] + u[2] + u[3]);
#pragma unroll
    for (int o = 16; o > 0; o >>= 1) cnt += __shfl_down(cnt, o, 32);
    if ((threadIdx.x & (warpSize - 1)) == 0)
        atomicAdd(&counts[b], cnt);
}

// ---------------------------------------------------------------------------
// Kernel 2: per-(sample, array) exact top-k sum via 4-pass radix select on the
// positive-float bit patterns, then one sum pass. One 1024-thread block per
// task; each scan streams the 1 MB array through LDS with a double-buffered
// TDM pipeline (wave 0 issues tensor_load_to_lds and waits TENSORcnt).
// ---------------------------------------------------------------------------
__global__ __launch_bounds__(SEL_T) void ohem_select_kernel(
    const float* __restrict__ d_all,   // [2 * BATCH * NPIX] : dA then dC
    const int*   __restrict__ counts,  // [BATCH]
    float*       __restrict__ results) // [2 * BATCH]
{
    __shared__ float    buf[2][CHUNK];
    __shared__ int      hist[256];
    __shared__ unsigned sPref;
    __shared__ int      sKK;
    __shared__ float    partial[SEL_T / 32];

    const int tid  = threadIdx.x;
    const int task = blockIdx.x;            // 0..15
    const int b    = task >> 1;
    const float* src = d_all + (size_t)(task & 1) * ((size_t)BATCH * NPIX)
                             + (size_t)b * NPIX;

    const int cnt = counts[b];
    const int k   = (int)floorf((float)cnt * RADIO);
    if (k <= 0) {                           // uniform per block
        if (tid == 0) results[task] = 0.0f;
        return;
    }

    if (tid == 0) { sPref = 0u; sKK = k; }
    if (tid < 256) hist[tid] = 0;
    __syncthreads();

    const int  nchunks = NPIX / CHUNK;      // 64
    const bool issuer  = (tid < 32);        // wave 0 drives the TDM

    // ---- 4 radix-select passes over bytes [31:24] .. [7:0] ----
    for (int pass = 0; pass < 4; ++pass) {
        const int      shift = 24 - 8 * pass;
        const unsigned pref  = sPref;

        if (issuer) { tdm_load_f32_chunk(src, lds_byte_offset(buf[0]), CHUNK); tdm_wait_all(); }
        __syncthreads();
        for (int c = 0; c < nchunks; ++c) {
            const int cur = c & 1;
            if (issuer && (c + 1) < nchunks)
                tdm_load_f32_chunk(src + (size_t)(c + 1) * CHUNK,
                                   lds_byte_offset(buf[(c + 1) & 1]), CHUNK);
            for (int i = tid; i < CHUNK; i += SEL_T) {
                const unsigned bits = __float_as_uint(buf[cur][i]);
                const bool act = (pass == 0) ? true
                                             : (((bits ^ pref) >> (shift + 8)) == 0u);
                if (act) atomicAdd(&hist[(bits >> shift) & 255u], 1);
            }
            if (issuer && (c + 1) < nchunks) tdm_wait_all();
            __syncthreads();
        }
        if (tid == 0) {                      // scan bins high -> low
            const int kk = sKK;
            int acc = 0, j = 255;
            for (; j >= 0; --j) { acc += hist[j]; if (acc >= kk) break; }
            if (j < 0) j = 0;
            sKK   = kk - (acc - hist[j]);    // residual rank inside chosen bin
            sPref = pref | ((unsigned)j << shift);
        }
        __syncthreads();
        if (tid < 256) hist[tid] = 0;
        __syncthreads();
    }

    const unsigned T  = sPref;               // bits of the k-th largest value
    const int      kk = sKK;                 // how many copies of T to include

    // ---- sum pass: strictly-greater values + kk * T ----
    float lsum = 0.0f;
    if (issuer) { tdm_load_f32_chunk(src, lds_byte_offset(buf[0]), CHUNK); tdm_wait_all(); }
    __syncthreads();
    for (int c = 0; c < nchunks; ++c) {
        const int cur = c & 1;
        if (issuer && (c + 1) < nchunks)
            tdm_load_f32_chunk(src + (size_t)(c + 1) * CHUNK,
                               lds_byte_offset(buf[(c + 1) & 1]), CHUNK);
        for (int i = tid; i < CHUNK; i += SEL_T) {
            const float v = buf[cur][i];
            if (__float_as_uint(v) > T) lsum += v;
        }
        if (issuer && (c + 1) < nchunks) tdm_wait_all();
        __syncthreads();
    }
    // deterministic tree reduction (wave32 shuffles)
    for (int o = 16; o > 0; o >>= 1) lsum += __shfl_down(lsum, o, 32);
    if ((tid & 31) == 0) partial[tid >> 5] = lsum;
    __syncthreads();
    if (tid < 32) {
        float v = partial[tid];
        for (int o = 16; o > 0; o >>= 1) v += __shfl_down(v, o, 32);
        if (tid == 0)
            results[task] = (v + (float)kk * __uint_as_float(T)) / ((float)k + EPS_F);
    }
}

// ---------------------------------------------------------------------------
// Kernel 3: out = 0.5*mean_b(alpha) + 0.5*mean_b(comp) = sum(results)/16
// ---------------------------------------------------------------------------
__global__ void ohem_final_kernel(const float* __restrict__ results,
                                  float* __restrict__ out)
{
    if (threadIdx.x == 0) {
        float s = 0.0f;
        for (int t = 0; t < 2 * BATCH; ++t) s += results[t];
        out[0] = s * (1.0f / 16.0f);
    }
}

// ---------------------------------------------------------------------------
extern "C" void kernel_launch(void* const* d_in, const int* in_sizes, int n_in,
                              void* d_out, int out_size, void* d_ws, size_t ws_size,
                              hipStream_t stream)
{
    (void)in_sizes; (void)n_in; (void)out_size; (void)ws_size;

    const float4* image  = (const float4*)d_in[0];
    const float4* alpha  = (const float4*)d_in[1];
    const float4* pred   = (const float4*)d_in[2];
    const int4*   trimap = (const int4*)  d_in[3];
    const float4* fg     = (const float4*)d_in[4];
    const float4* bg     = (const float4*)d_in[5];
    float*        out    = (float*)d_out;

    // workspace layout: dA[B*N] | dC[B*N] | counts[64 ints] | results[16 f32]
    float* dA      = (float*)d_ws;
    float* dC      = dA + (size_t)BATCH * NPIX;
    int*   counts  = (int*)(dC + (size_t)BATCH * NPIX);
    float* results = (float*)(counts + 64);

    ohem_init_kernel<<<1, 64, 0, stream>>>(counts);
    ohem_diff_kernel<<<(BATCH * NPIX / 4) / K1_T, K1_T, 0, stream>>>(
        image, alpha, pred, trimap, fg, bg,
        (float4*)dA, (float4*)dC, counts);
    ohem_select_kernel<<<2 * BATCH, SEL_T, 0, stream>>>(dA, counts, results);
    ohem_final_kernel<<<1, 32, 0, stream>>>(results, out);
}